// OSM_9938554322991
// MI455X (gfx1250) — compile-verified
//
#include <hip/hip_runtime.h>
#include <hip/hip_bf16.h>
#include <math.h>

// ---------------- problem constants ----------------
#define NN        512
#define CC        2048
#define HW        128
#define NCLS      751
#define NCLS_PAD  752
#define ALPHA_    1.2f
#define LAMB_     0.5f
#define SIGMA2_   0.64f     // 0.8^2
#define BN_EPS_   1e-5f

typedef float v2f __attribute__((ext_vector_type(2)));
typedef float v4f __attribute__((ext_vector_type(4)));
typedef float v8f __attribute__((ext_vector_type(8)));

// ---------------- workspace layout (floats) ----------------
#define WS_MU      0                          // [CC]
#define WS_SCALE   (WS_MU + CC)               // [CC]
#define WS_BNN     (WS_SCALE + CC)            // [NN*CC]
#define WS_WN      (WS_BNN + NN*CC)           // [NCLS*CC]
#define WS_G       (WS_WN + NCLS*CC)          // [NN*NN]
#define WS_LOGIT   (WS_G + NN*NN)             // [NN*NCLS_PAD]
#define WS_ATTEN   (WS_LOGIT + NN*NCLS_PAD)   // [NN]
#define WS_ACCUM   (WS_ATTEN + NN)            // [4]

// ---------------- reductions (wave32) ----------------
__device__ __forceinline__ float waveSum(float v) {
#pragma unroll
    for (int o = 16; o > 0; o >>= 1) v += __shfl_xor(v, o, 32);
    return v;
}
__device__ __forceinline__ float waveMax(float v) {
#pragma unroll
    for (int o = 16; o > 0; o >>= 1) v = fmaxf(v, __shfl_xor(v, o, 32));
    return v;
}
__device__ float blockSum(float v) {
    __shared__ float s[8];
    int lane = threadIdx.x & 31, wid = threadIdx.x >> 5;
    v = waveSum(v);
    if (lane == 0) s[wid] = v;
    __syncthreads();
    float r;
    if (wid == 0) {
        r = (lane < (int)(blockDim.x >> 5)) ? s[lane] : 0.0f;
        r = waveSum(r);
        if (lane == 0) s[0] = r;
    }
    __syncthreads();
    r = s[0];
    __syncthreads();
    return r;
}
__device__ float blockMax(float v) {
    __shared__ float s[8];
    int lane = threadIdx.x & 31, wid = threadIdx.x >> 5;
    v = waveMax(v);
    if (lane == 0) s[wid] = v;
    __syncthreads();
    float r;
    if (wid == 0) {
        r = (lane < (int)(blockDim.x >> 5)) ? s[lane] : -INFINITY;
        r = waveMax(r);
        if (lane == 0) s[0] = r;
    }
    __syncthreads();
    r = s[0];
    __syncthreads();
    return r;
}

// ---------------- 1) global average pool: one wave per (n,c) ----------------
// Streams 512 MB read-once: non-temporal loads so the 192 MB L2 isn't
// polluted; the 4 MB global_feat result stays regular-temporal (reused).
__global__ void osm_pool(const float* __restrict__ feat, float* __restrict__ gf) {
    int wave = (blockIdx.x * (blockDim.x >> 5)) + (threadIdx.x >> 5); // row = n*CC + c
    int lane = threadIdx.x & 31;
    const v4f* p = (const v4f*)(feat + (size_t)wave * HW);
    v4f v = __builtin_nontemporal_load(p + lane);   // 32 lanes x 16B = 512B/row
    float s = waveSum(v.x + v.y + v.z + v.w);
    if (lane == 0) gf[wave] = s * (1.0f / (float)HW);
}

// ---------------- 2) BN stats per channel ----------------
__global__ void osm_bnstats(const float* __restrict__ gf, const float* __restrict__ gamma,
                            float* __restrict__ ws) {
    int c = blockIdx.x;
    float s = 0.f, ss = 0.f;
    for (int n = threadIdx.x; n < NN; n += blockDim.x) {
        float x = gf[(size_t)n * CC + c];
        s += x; ss += x * x;
    }
    s  = blockSum(s);
    ss = blockSum(ss);
    if (threadIdx.x == 0) {
        float mu  = s * (1.0f / NN);
        float var = ss * (1.0f / NN) - mu * mu;
        ws[WS_MU + c]    = mu;
        ws[WS_SCALE + c] = gamma[c] * rsqrtf(var + BN_EPS_);
    }
}

// ---------------- 3) per-row BN apply + L2 normalize -> bnn ----------------
__global__ void osm_rownorm(const float* __restrict__ gf, float* __restrict__ ws) {
    int n = blockIdx.x;
    const float* mu = ws + WS_MU;
    const float* sc = ws + WS_SCALE;
    float* bnn = ws + WS_BNN;
    float vals[CC / 256];
    float ss = 0.f;
#pragma unroll
    for (int j = 0; j < CC / 256; ++j) {
        int c = threadIdx.x + j * 256;
        float v = (gf[(size_t)n * CC + c] - mu[c]) * sc[c];
        vals[j] = v; ss += v * v;
    }
    float tot = blockSum(ss);
    float rinv = 1.0f / fmaxf(sqrtf(tot), 1e-12f);
#pragma unroll
    for (int j = 0; j < CC / 256; ++j) {
        int c = threadIdx.x + j * 256;
        bnn[(size_t)n * CC + c] = vals[j] * rinv;
    }
}

// ---------------- 4) L2 normalize classifier rows -> wn ----------------
__global__ void osm_wnorm(const float* __restrict__ w, float* __restrict__ ws) {
    int r = blockIdx.x;
    float* wn = ws + WS_WN;
    float vals[CC / 256];
    float ss = 0.f;
#pragma unroll
    for (int j = 0; j < CC / 256; ++j) {
        int c = threadIdx.x + j * 256;
        float v = w[(size_t)r * CC + c];
        vals[j] = v; ss += v * v;
    }
    float tot = blockSum(ss);
    float rinv = 1.0f / fmaxf(sqrtf(tot), 1e-12f);
#pragma unroll
    for (int j = 0; j < CC / 256; ++j) {
        int c = threadIdx.x + j * 256;
        wn[(size_t)r * CC + c] = vals[j] * rinv;
    }
}

// ---------------- 5) Gram matrix G = bnn * bnn^T via V_WMMA_F32_16X16X4_F32 ----
// One wave per 16x16 output tile. A-tile 16x4 / B-tile 4x16 per ISA layout:
// lanes 0-15 hold K={0,1}, lanes 16-31 hold K={2,3}; M/N = lane&15.
__global__ void osm_gram(const float* __restrict__ ws_bnn, float* __restrict__ G) {
    int wave = (blockIdx.x * (blockDim.x >> 5)) + (threadIdx.x >> 5);
    int lane = threadIdx.x & 31;
    int ti = wave >> 5, tj = wave & 31;           // 32 x 32 tiles
    int i0 = ti * 16, j0 = tj * 16;
    int m = lane & 15, half = lane >> 4;
    const float* arow = ws_bnn + (size_t)(i0 + m) * CC + 2 * half;
    const float* brow = ws_bnn + (size_t)(j0 + m) * CC + 2 * half;
    v8f c = {};
#pragma unroll 4
    for (int k = 0; k < CC; k += 4) {
        v2f a = *(const v2f*)(arow + k);
        v2f b = *(const v2f*)(brow + k);
        c = __builtin_amdgcn_wmma_f32_16x16x4_f32(false, a, false, b,
                                                  (short)0, c, false, false);
    }
    int col = j0 + m;
    float* out = G + (size_t)(i0 + half * 8) * NN + col;
#pragma unroll
    for (int r = 0; r < 8; ++r) out[(size_t)r * NN] = c[r];
}

// ---------------- 6) logits = bnn * wn^T via WMMA, N padded to 752 ------------
__global__ void osm_logits(const float* __restrict__ ws_bnn, const float* __restrict__ ws_wn,
                           float* __restrict__ logits) {
    int wave = (blockIdx.x * (blockDim.x >> 5)) + (threadIdx.x >> 5);
    int lane = threadIdx.x & 31;
    const int TN = NCLS_PAD / 16;                 // 47 tiles in class dim
    int tm = wave / TN, tn = wave % TN;
    int i0 = tm * 16, j0 = tn * 16;
    int m = lane & 15, half = lane >> 4;
    int jr = j0 + m;
    float bmask = (jr < NCLS) ? 1.0f : 0.0f;      // convergent mask, no branch
    int jrc = jr < NCLS ? jr : (NCLS - 1);
    const float* arow = ws_bnn + (size_t)(i0 + m) * CC + 2 * half;
    const float* brow = ws_wn  + (size_t)jrc * CC + 2 * half;
    v8f c = {};
#pragma unroll 4
    for (int k = 0; k < CC; k += 4) {
        v2f a = *(const v2f*)(arow + k);
        v2f b = *(const v2f*)(brow + k);
        b *= bmask;
        c = __builtin_amdgcn_wmma_f32_16x16x4_f32(false, a, false, b,
                                                  (short)0, c, false, false);
    }
    int col = j0 + m;
    if (col < NCLS) {
        float* out = logits + (size_t)(i0 + half * 8) * NCLS_PAD + col;
#pragma unroll
        for (int r = 0; r < 8; ++r) out[(size_t)r * NCLS_PAD] = c[r];
    }
}

// ---------------- 7) softmax per row -> atten[n] = probs[n, target[n]] --------
__global__ void osm_atten(const float* __restrict__ logits, const int* __restrict__ targets,
                          float* __restrict__ atten) {
    int n = blockIdx.x;
    const float* row = logits + (size_t)n * NCLS_PAD;
    float mx = -INFINITY;
    for (int c = threadIdx.x; c < NCLS; c += blockDim.x) mx = fmaxf(mx, row[c]);
    mx = blockMax(mx);
    float s = 0.f;
    for (int c = threadIdx.x; c < NCLS; c += blockDim.x) s += expf(row[c] - mx);
    s = blockSum(s);
    if (threadIdx.x == 0) {
        int t = targets[n];
        atten[n] = expf(row[t] - mx) / s;
    }
}

// ---------------- 8) zero the 4 accumulators ----------------
__global__ void osm_init(float* __restrict__ acc) {
    if (threadIdx.x < 4) acc[threadIdx.x] = 0.0f;
}

// ---------------- 9) pairwise reduction over 512x512 ----------------
__global__ void osm_pairs(const float* __restrict__ G, const float* __restrict__ atten,
                          const int* __restrict__ targets, float* __restrict__ acc) {
    float sWP = 0.f, sWN = 0.f, sWPd = 0.f, sWNs = 0.f;
    int stride = gridDim.x * blockDim.x;
    for (int idx = blockIdx.x * blockDim.x + threadIdx.x; idx < NN * NN; idx += stride) {
        int i = idx >> 9, j = idx & (NN - 1);
        float sq_i = G[(size_t)i * NN + i];
        float sq_j = G[(size_t)j * NN + j];
        float gij  = G[idx];
        float d2   = fmaxf(sq_i + sq_j - 2.0f * gij, 1e-12f); // == dist*dist in ref
        float dist = sqrtf(d2);
        float sneg = fmaxf(ALPHA_ - dist, 1e-12f);
        float spos = expf(-d2 / SIGMA2_);
        bool  p    = (targets[i] == targets[j]);
        float A    = fminf(atten[i], atten[j]);
        float offd = (i != j) ? 1.0f : 0.0f;
        float w    = (p ? spos : sneg) * A * offd;
        if (p) { sWP += w; sWPd += w * d2; }
        else   { sWN += w; sWNs += w * sneg * sneg; }
    }
    sWP  = blockSum(sWP);
    sWN  = blockSum(sWN);
    sWPd = blockSum(sWPd);
    sWNs = blockSum(sWNs);
    if (threadIdx.x == 0) {
        atomicAdd(&acc[0], sWP);
        atomicAdd(&acc[1], sWN);
        atomicAdd(&acc[2], sWPd);
        atomicAdd(&acc[3], sWNs);
    }
}

// ---------------- 10) finalize scalar loss ----------------
__global__ void osm_final(const float* __restrict__ acc, float* __restrict__ out) {
    if (threadIdx.x == 0 && blockIdx.x == 0) {
        float LP = 0.5f * acc[2] / acc[0];
        float LN = 0.5f * acc[3] / acc[1];
        out[0] = (1.0f - LAMB_) * LP + LAMB_ * LN;
    }
}

extern "C" void kernel_launch(void* const* d_in, const int* in_sizes, int n_in,
                              void* d_out, int out_size, void* d_ws, size_t ws_size,
                              hipStream_t stream) {
    const float* features = (const float*)d_in[0];
    const int*   targets  = (const int*)d_in[1];
    const float* gamma    = (const float*)d_in[2];
    const float* weight   = (const float*)d_in[3];

    float* out = (float*)d_out;          // out[0] = L, out[1..] = global_feat [512,2048]
    float* gf  = out + 1;
    float* ws  = (float*)d_ws;

    // 1) pool: one wave per (n,c); 8 waves per 256-thread block (NT streaming)
    osm_pool<<<(NN * CC) / 8, 256, 0, stream>>>(features, gf);
    // 2) BN stats per channel
    osm_bnstats<<<CC, 256, 0, stream>>>(gf, gamma, ws);
    // 3) BN apply + row L2 normalize
    osm_rownorm<<<NN, 256, 0, stream>>>(gf, ws);
    // 4) classifier row L2 normalize
    osm_wnorm<<<NCLS, 256, 0, stream>>>(weight, ws);
    // 5) Gram matrix (1024 tiles, 8 waves/block)
    osm_gram<<<(32 * 32) / 8, 256, 0, stream>>>(ws + WS_BNN, ws + WS_G);
    // 6) logits (32*47 = 1504 tiles)
    osm_logits<<<(32 * (NCLS_PAD / 16)) / 8, 256, 0, stream>>>(ws + WS_BNN, ws + WS_WN,
                                                               ws + WS_LOGIT);
    // 7) softmax -> atten
    osm_atten<<<NN, 256, 0, stream>>>(ws + WS_LOGIT, targets, ws + WS_ATTEN);
    // 8) zero accumulators
    osm_init<<<1, 32, 0, stream>>>(ws + WS_ACCUM);
    // 9) pairwise reduction
    osm_pairs<<<256, 256, 0, stream>>>(ws + WS_G, ws + WS_ATTEN, targets, ws + WS_ACCUM);
    // 10) finalize
    osm_final<<<1, 32, 0, stream>>>(ws + WS_ACCUM, out);
}